// XENON_GCNN_74440373174758
// MI455X (gfx1250) — compile-verified
//
#include <hip/hip_runtime.h>
#include <hip/hip_bf16.h>

#define FINAL_OUT 28
#define NPG 127
#define NB  4096
#define NTOT (NB * NPG)
#define EPG 2032
#define ETOT (NB * EPG)
#define GFEAT (FINAL_OUT * NPG)   // 3556

typedef float v2f __attribute__((ext_vector_type(2)));
typedef float v8f __attribute__((ext_vector_type(8)));
typedef unsigned u32x4 __attribute__((ext_vector_type(4)));
typedef int i32x4 __attribute__((ext_vector_type(4)));
typedef int i32x8 __attribute__((ext_vector_type(8)));

#if __has_builtin(__builtin_amdgcn_tensor_load_to_lds) && \
    __has_builtin(__builtin_amdgcn_s_wait_tensorcnt)
#define USE_TDM 1
#else
#define USE_TDM 0
#endif

#if USE_TDM
// 1-D contiguous copy of `nwords` u32 from global to LDS via the Tensor Data
// Mover.  D# layout per CDNA5 ISA ch.8: group0 = {count|flags, lds_addr,
// global_addr lo, global_addr hi | type=2}; group1 holds data_size=4B,
// tensor_dim0 = nwords, degenerate dim1 = 1, tile = (nwords, 1).
// This toolchain's builtin takes 6 args (g0, g1, g2, g3, extra, cpol).
__device__ __forceinline__ void tdm_copy_u32_to_lds(const void* gptr,
                                                    unsigned lds_addr,
                                                    int nwords) {
    unsigned long long ga = (unsigned long long)(size_t)gptr;
    u32x4 g0;
    g0[0] = 1u;                                        // count=1 (user D#)
    g0[1] = lds_addr;                                  // lds_addr [63:32]
    g0[2] = (unsigned)(ga & 0xFFFFFFFFull);            // global_addr[31:0]
    g0[3] = (unsigned)((ga >> 32) & 0x1FFFFFFull)      // global_addr[56:32]
            | (2u << 30);                              // type=2 ("image")
    i32x8 g1;
    g1[0] = (int)(2u << 16);                           // data_size=2 (4B)
    g1[1] = (int)(((unsigned)nwords & 0xFFFFu) << 16); // tensor_dim0[15:0]
    g1[2] = (int)((((unsigned)nwords >> 16) & 0xFFFFu) // tensor_dim0[31:16]
            | (1u << 16));                             // tensor_dim1 = 1
    g1[3] = (int)(((unsigned)nwords & 0xFFFFu) << 16); // tile_dim0 = nwords
    g1[4] = 1;                                         // tile_dim1=1, dim2=0
    g1[5] = nwords;                                    // dim0_stride[31:0]
    g1[6] = 0;
    g1[7] = 0;
    i32x4 z4 = {0, 0, 0, 0};                           // groups 2/3 unused
    i32x8 z8 = {0, 0, 0, 0, 0, 0, 0, 0};
    __builtin_amdgcn_tensor_load_to_lds(g0, g1, z4, z4, z8, 0);
}
#endif

// ---------------------------------------------------------------------------
// Kernel 1: fully fused per-graph GCN (degree, layer1, layer2) in LDS.
// One 256-thread block (8 waves) per graph. WMMA f32 16x16x4 for X@W1, H@W2.
// dinv row-scaling is folded into the WMMA A fragments; edge scatters are
// edge-major (ds_load_b128 source rows + ds_add_f32 atomics).
// ---------------------------------------------------------------------------
__global__ __launch_bounds__(256) void gcn_fused_kernel(
    const float* __restrict__ x,
    const int* __restrict__ esrc,
    const int* __restrict__ edst,
    const float* __restrict__ W1, const float* __restrict__ b1,
    const float* __restrict__ W2, const float* __restrict__ b2,
    float* __restrict__ g_out)
{
    __shared__ float    s_x[128 * 4];        // node features, K padded 3->4
    __shared__ float    s_w1[4 * 16];        // W1 padded K 3->4
    __shared__ float    s_w2[16 * 32];       // W2 padded N 28->32
    __shared__ unsigned s_edges[EPG];        // (src<<16)|dst, local ids
    __shared__ int      s_deg[128];
    __shared__ float    s_dinv[128];
    __shared__ float    s_t[128 * 16];       // TDM stage A, then t1 / h1
    __shared__ float    s_acc[128 * 16];     // TDM stage B, then layer1 acc
    __shared__ float    s_t2[128 * 32];      // t2, feature dim padded to 32
    __shared__ float    s_acc2[128 * 32];    // layer2 acc, padded to 32

    const int g    = blockIdx.x;
    const int tid  = threadIdx.x;
    const int lane = tid & 31;
    const int wave = tid >> 5;
    const int base = g * NPG;
    const size_t ebase = (size_t)g * EPG;

#if USE_TDM
    // Wave 0 kicks off the async edge-list DMA into LDS staging buffers.
    if (wave == 0) {
        tdm_copy_u32_to_lds(esrc + ebase, (unsigned)(size_t)(void*)s_t, EPG);
        tdm_copy_u32_to_lds(edst + ebase, (unsigned)(size_t)(void*)s_acc, EPG);
    }
#else
    __builtin_prefetch(&esrc[ebase + tid], 0, 1);
    __builtin_prefetch(&edst[ebase + tid], 0, 1);
#endif

    // ---- Phase A: weight / feature loads + init (overlaps the DMA) ----
    for (int i = tid; i < 128; i += 256) s_deg[i] = 1;          // self-loop
    for (int i = tid; i < 64; i += 256) s_w1[i] = (i < 48) ? W1[i] : 0.0f;
    for (int i = tid; i < 16 * 32; i += 256) {
        int k = i >> 5, n = i & 31;
        s_w2[i] = (n < FINAL_OUT) ? W2[k * FINAL_OUT + n] : 0.0f;
    }
    for (int node = tid; node < 128; node += 256) {
        const float* xr = x + (size_t)(base + node) * 3;
        bool v = node < NPG;
        s_x[node * 4 + 0] = v ? xr[0] : 0.0f;
        s_x[node * 4 + 1] = v ? xr[1] : 0.0f;
        s_x[node * 4 + 2] = v ? xr[2] : 0.0f;
        s_x[node * 4 + 3] = 0.0f;
    }

#if USE_TDM
    if (wave == 0) __builtin_amdgcn_s_wait_tensorcnt(0);
#endif
    __syncthreads();

    // ---- pack edges: (src<<16)|dst with graph-local node ids ----
    {
#if USE_TDM
        const int* stA = (const int*)s_t;
        const int* stB = (const int*)s_acc;
#endif
        for (int e = tid; e < EPG; e += 256) {
#if USE_TDM
            int s = stA[e] - base;
            int d = stB[e] - base;
#else
            int s = esrc[ebase + e] - base;
            int d = edst[ebase + e] - base;
#endif
            s_edges[e] = ((unsigned)s << 16) | (unsigned)d;
        }
    }
    __syncthreads();

    // ---- degree + dinv ----
    for (int e = tid; e < EPG; e += 256)
        atomicAdd(&s_deg[s_edges[e] & 0xFFFFu], 1);
    __syncthreads();
    for (int i = tid; i < 128; i += 256)
        s_dinv[i] = rsqrtf((float)s_deg[i]);
    __syncthreads();

    // ---- layer 1 dense: t1 = (dinv ⊙ X) @ W1 via WMMA f32 16x16x4 ----
    {
        const int m  = lane & 15;
        const int hi = lane >> 4;
        const int kh = hi << 1;          // 0 or 2 (K half per lane group)
        const int n  = lane & 15;
        const int row = wave * 16 + m;
        const float dv = s_dinv[row];
        v2f a, bf;
        a.x = s_x[row * 4 + kh] * dv;
        a.y = s_x[row * 4 + kh + 1] * dv;
        bf.x = s_w1[kh * 16 + n];
        bf.y = s_w1[(kh + 1) * 16 + n];
        v8f c = {};
        v8f d = __builtin_amdgcn_wmma_f32_16x16x4_f32(
            false, a, false, bf, (short)0, c, false, false);
#pragma unroll
        for (int r = 0; r < 8; ++r) {
            int node = wave * 16 + r + hi * 8;
            s_t[node * 16 + n]   = d[r];   // t1 = dinv[src]*h[src]
            s_acc[node * 16 + n] = d[r];   // self-loop seed
        }
    }
    __syncthreads();

    // ---- layer-1 edge scatter: edge-major, b128 source loads ----
#pragma clang loop unroll(disable)
    for (int e = tid; e < EPG; e += 256) {
        unsigned pk = s_edges[e];
        int s = (int)(pk >> 16), d = (int)(pk & 0xFFFFu);
        const float4* ts = (const float4*)&s_t[s * 16];
        float* ad = &s_acc[d * 16];
#pragma unroll
        for (int q = 0; q < 4; ++q) {
            float4 v = ts[q];                 // ds_load_b128
            atomicAdd(&ad[q * 4 + 0], v.x);
            atomicAdd(&ad[q * 4 + 1], v.y);
            atomicAdd(&ad[q * 4 + 2], v.z);
            atomicAdd(&ad[q * 4 + 3], v.w);
        }
    }
    __syncthreads();

    // h1 = relu(dinv * acc + b1)  (stored back into s_t; pad rows harmless)
    for (int i = tid; i < 128 * 16; i += 256) {
        int node = i >> 4;
        int f = i & 15;
        s_t[i] = fmaxf(s_dinv[node] * s_acc[i] + b1[f], 0.0f);
    }
    __syncthreads();

    // ---- layer 2 dense: t2 = (dinv ⊙ H1) @ W2 (K=16 -> 4 WMMAs, 2 N tiles)
    {
        const int m  = lane & 15;
        const int hi = lane >> 4;
        const int n  = lane & 15;
        const int row = wave * 16 + m;
        const float dv = s_dinv[row];
#pragma unroll
        for (int nt = 0; nt < 2; ++nt) {
            v8f c = {};
#pragma unroll
            for (int kk = 0; kk < 4; ++kk) {
                int kh = kk * 4 + (hi << 1);
                v2f a, bf;
                a.x = s_t[row * 16 + kh] * dv;
                a.y = s_t[row * 16 + kh + 1] * dv;
                bf.x = s_w2[kh * 32 + nt * 16 + n];
                bf.y = s_w2[(kh + 1) * 32 + nt * 16 + n];
                c = __builtin_amdgcn_wmma_f32_16x16x4_f32(
                    false, a, false, bf, (short)0, c, false, false);
            }
            int f = nt * 16 + n;
#pragma unroll
            for (int r = 0; r < 8; ++r) {
                int node = wave * 16 + r + hi * 8;
                s_t2[node * 32 + f]   = c[r];
                s_acc2[node * 32 + f] = c[r];
            }
        }
    }
    __syncthreads();

    // ---- layer-2 edge scatter: edge-major, 7x b128 loads + 28 atomics ----
#pragma clang loop unroll(disable)
    for (int e = tid; e < EPG; e += 256) {
        unsigned pk = s_edges[e];
        int s = (int)(pk >> 16), d = (int)(pk & 0xFFFFu);
        const float4* ts = (const float4*)&s_t2[s * 32];
        float* ad = &s_acc2[d * 32];
#pragma unroll
        for (int q = 0; q < 7; ++q) {         // 28 floats = 7 x float4
            float4 v = ts[q];                 // ds_load_b128
            atomicAdd(&ad[q * 4 + 0], v.x);
            atomicAdd(&ad[q * 4 + 1], v.y);
            atomicAdd(&ad[q * 4 + 2], v.z);
            atomicAdd(&ad[q * 4 + 3], v.w);
        }
    }
    __syncthreads();

    // ---- g = relu(dinv * acc2 + b2), float4 stores (global_store_b128) ----
    float4* gout4 = (float4*)(g_out + (size_t)g * GFEAT);
#pragma clang loop unroll(disable)
    for (int i4 = tid; i4 < GFEAT / 4; i4 += 256) {    // 889 float4s
        float4 o;
#pragma unroll
        for (int c = 0; c < 4; ++c) {
            int p = i4 * 4 + c;
            int node = p / FINAL_OUT;
            int f = p - node * FINAL_OUT;
            float v = fmaxf(s_dinv[node] * s_acc2[node * 32 + f] + b2[f], 0.0f);
            ((float*)&o)[c] = v;
        }
        gout4[i4] = o;
    }
}

// ---------------------------------------------------------------------------
// Kernel 2: head — out = relu(G @ lw1 + lb1) @ lw2 + lb2
// G: [4096, 3556]. One wave per 16-row tile; 3556 = 28*127, so the K chain is
// 127 rolled iterations of 7 WMMAs each (kept rolled for I$).
// ---------------------------------------------------------------------------
__global__ __launch_bounds__(256) void head_kernel(
    const float* __restrict__ gmat,
    const float* __restrict__ lw1, const float* __restrict__ lb1,
    const float* __restrict__ lw2, const float* __restrict__ lb2,
    float* __restrict__ out)
{
    __shared__ float s_H[8][16][17];   // per-wave relu(G@lw1) tile, padded

    const int tid  = threadIdx.x;
    const int lane = tid & 31;
    const int wave = tid >> 5;
    const int tile = blockIdx.x * 8 + wave;      // 0..255

    const int m  = lane & 15;
    const int hi = lane >> 4;
    const int kh = hi << 1;
    const int n  = lane & 15;
    const int row = tile * 16 + m;
    const v2f* arow2 = (const v2f*)(gmat + (size_t)row * GFEAT);

    v8f c = {};
#pragma clang loop unroll(disable)
    for (int kb = 0; kb < GFEAT; kb += 28) {
#pragma unroll
        for (int kk = 0; kk < 7; ++kk) {
            int k0 = kb + kk * 4 + kh;           // always even
            v2f a = arow2[k0 >> 1];              // global_load_b64
            v2f bf;
            bf.x = lw1[(size_t)k0 * 16 + n];
            bf.y = lw1[(size_t)(k0 + 1) * 16 + n];
            c = __builtin_amdgcn_wmma_f32_16x16x4_f32(
                false, a, false, bf, (short)0, c, false, false);
        }
    }
    float bn = lb1[n];
#pragma unroll
    for (int r = 0; r < 8; ++r)
        s_H[wave][r + hi * 8][n] = fmaxf(c[r] + bn, 0.0f);
    __syncthreads();

    // tiny epilogue: [16,16] @ [16,2]; each lane -> one (row, j)
    int m2 = lane & 15;
    int j  = lane >> 4;          // 0 or 1
    float acc = lb2[j];
#pragma unroll
    for (int nn = 0; nn < 16; ++nn)
        acc += s_H[wave][m2][nn] * lw2[nn * 2 + j];
    out[(size_t)(tile * 16 + m2) * 2 + j] = acc;
}

// ---------------------------------------------------------------------------
extern "C" void kernel_launch(void* const* d_in, const int* in_sizes, int n_in,
                              void* d_out, int out_size, void* d_ws, size_t ws_size,
                              hipStream_t stream) {
    const float* x    = (const float*)d_in[0];
    const int*   eidx = (const int*)d_in[1];   // [2, E] int32
    const float* W1   = (const float*)d_in[2];
    const float* b1   = (const float*)d_in[3];
    const float* W2   = (const float*)d_in[4];
    const float* b2   = (const float*)d_in[5];
    const float* lw1  = (const float*)d_in[6];
    const float* lb1  = (const float*)d_in[7];
    const float* lw2  = (const float*)d_in[8];
    const float* lb2  = (const float*)d_in[9];
    float* out  = (float*)d_out;
    float* gbuf = (float*)d_ws;                // NB * 3556 floats = 58.3 MB

    const int* esrc = eidx;
    const int* edst = eidx + (size_t)ETOT;

    gcn_fused_kernel<<<NB, 256, 0, stream>>>(x, esrc, edst, W1, b1, W2, b2, gbuf);
    head_kernel<<<NB / 128, 256, 0, stream>>>(gbuf, lw1, lb1, lw2, lb2, out);
}